// Part_similarity_constrain_46462956208692
// MI455X (gfx1250) — compile-verified
//
#include <hip/hip_runtime.h>
#include <math.h>

#define B 128
#define D 2048
#define P 6
#define NPAIR 15
#define NCOL 21            // 6 xn columns + 15 sqrt(2)-scaled pair-diff columns
#define K (NCOL * D)       // 43008 features per sample
#define SPLIT 16           // split-K factor for the Gram
#define KS (K / SPLIT)     // 2688 (multiple of 4)
#define TILES 64           // 8x8 tiles of 16x16 over the 128x128 Gram

typedef float v2f __attribute__((ext_vector_type(2)));
typedef float v4f __attribute__((ext_vector_type(4)));
typedef float v8f __attribute__((ext_vector_type(8)));

__constant__ int c_pair_p[NPAIR] = {0,0,0,0,0,1,1,1,1,2,2,2,3,3,4};
__constant__ int c_pair_q[NPAIR] = {1,2,3,4,5,2,3,4,5,3,4,5,4,5,5};

// ---------------------------------------------------------------------------
// Kernel 1: per-sample feature build + lambda, fused.
// One block per b. Stage contiguous x[b] (2048*6 floats, 48KB) in LDS via
// coalesced float4 loads, then:
//   - 21 block reductions: 6 sumsq(x_p) + 15 dot(x_p,x_q)
//   - diff norms from identity ||x_p-x_q||^2 = s_p + s_q - 2*dot_pq
//   - write 6 unit-scaled xn columns and 15 sqrt(2)-scaled diff columns to F
//   - lam[b] = 2/(P-1) * sum_{p<q} dot_pq / (max(n_p,eps)*max(n_q,eps))
// ---------------------------------------------------------------------------
__global__ void __launch_bounds__(256) build_features(const float* __restrict__ x,
                                                      float* __restrict__ F,
                                                      float* __restrict__ lam) {
    __shared__ float xbuf[D * P];     // 48 KB
    __shared__ float red[256];
    __shared__ float res[NCOL];       // 6 sumsq + 15 dots
    int b = blockIdx.x;
    int tid = threadIdx.x;

    // Coalesced load of x[b] into LDS: 3072 float4's, 12 per thread.
    {
        const v4f* src = (const v4f*)(x + (size_t)b * D * P);
        v4f* dst = (v4f*)xbuf;
        for (int i = tid; i < (D * P) / 4; i += 256) dst[i] = src[i];
    }
    __syncthreads();

    // 21 partial sums per thread over its d-slice (8 d's per thread).
    float acc[NCOL];
#pragma unroll
    for (int c = 0; c < NCOL; ++c) acc[c] = 0.f;
    for (int d = tid; d < D; d += 256) {
        float xv[P];
#pragma unroll
        for (int p = 0; p < P; ++p) xv[p] = xbuf[d * P + p];
#pragma unroll
        for (int p = 0; p < P; ++p) acc[p] += xv[p] * xv[p];
#pragma unroll
        for (int pr = 0; pr < NPAIR; ++pr)
            acc[P + pr] += xv[c_pair_p[pr]] * xv[c_pair_q[pr]];
    }
    // 21 sequential block reductions.
    for (int c = 0; c < NCOL; ++c) {
        red[tid] = acc[c];
        __syncthreads();
        for (int s = 128; s > 0; s >>= 1) {
            if (tid < s) red[tid] += red[tid + s];
            __syncthreads();
        }
        if (tid == 0) res[c] = red[0];
        __syncthreads();
    }

    // Scales (every thread computes from res; cheap).
    float n[P], sxn[P];
#pragma unroll
    for (int p = 0; p < P; ++p) {
        n[p]   = fmaxf(sqrtf(res[p]), 1e-12f);
        sxn[p] = 1.0f / n[p];
    }
    float sdiff[NPAIR];
#pragma unroll
    for (int pr = 0; pr < NPAIR; ++pr) {
        int p = c_pair_p[pr], q = c_pair_q[pr];
        float d2 = fmaxf(res[p] + res[q] - 2.0f * res[P + pr], 0.0f);
        sdiff[pr] = 1.41421356237309515f / fmaxf(sqrtf(d2), 1e-12f);
    }

    // Write F from LDS (columns contiguous in F).
    float* Fb = F + (size_t)b * K;
#pragma unroll
    for (int p = 0; p < P; ++p)
        for (int d = tid; d < D; d += 256)
            Fb[p * D + d] = xbuf[d * P + p] * sxn[p];
#pragma unroll
    for (int pr = 0; pr < NPAIR; ++pr) {
        int p = c_pair_p[pr], q = c_pair_q[pr];
        for (int d = tid; d < D; d += 256)
            Fb[(P + pr) * D + d] = (xbuf[d * P + p] - xbuf[d * P + q]) * sdiff[pr];
    }

    if (tid == 0) {
        float t = 0.f;
#pragma unroll
        for (int pr = 0; pr < NPAIR; ++pr)
            t += res[P + pr] * sxn[c_pair_p[pr]] * sxn[c_pair_q[pr]];
        lam[b] = 2.0f * t / (float)(P - 1);
    }
}

// ---------------------------------------------------------------------------
// Kernel 2: split-K Gram partials via V_WMMA_F32_16X16X4_F32.
// 1024 wave-jobs = 64 tiles x 16 K-slices; 4 waves per 128-thread block.
// A-fragment lane layout (ISA 7.12.2, 32-bit A 16x4):
//   lane<16:  {A[r][k0],   A[r][k0+1]},  lane>=16: {A[r][k0+2], A[r][k0+3]}
// B = A^T of the j row block -> identical per-lane load pattern on j rows.
// ---------------------------------------------------------------------------
__global__ void __launch_bounds__(128) gram_wmma(const float* __restrict__ F,
                                                 float* __restrict__ Gpart) {
    int job  = blockIdx.x * 4 + (threadIdx.x >> 5);   // 0..1023
    int tile = job >> 4;                              // 0..63
    int s    = job & (SPLIT - 1);                     // 0..15
    int ti = tile >> 3;
    int tj = tile & 7;
    int lane = threadIdx.x & 31;
    int r    = lane & 15;
    int koff = (lane >> 4) << 1;                      // 0 | 2
    const float* arow = F + (size_t)(ti * 16 + r) * K + (size_t)s * KS;
    const float* brow = F + (size_t)(tj * 16 + r) * K + (size_t)s * KS;
    v8f acc = {};
#pragma unroll 8
    for (int k0 = 0; k0 < KS; k0 += 4) {
        v2f a  = *(const v2f*)(arow + k0 + koff);
        v2f bv = *(const v2f*)(brow + k0 + koff);
        acc = __builtin_amdgcn_wmma_f32_16x16x4_f32(
            /*neg_a=*/false, a, /*neg_b=*/false, bv,
            /*c_mod=*/(short)0, acc, /*reuse_a=*/false, /*reuse_b=*/false);
    }
    // D layout: VGPR rr -> row rr (lanes 0-15) or rr+8 (lanes 16-31)
    int col   = tj * 16 + (lane & 15);
    int rbase = ti * 16 + ((lane >> 4) << 3);
    float* Gp = Gpart + (size_t)s * B * B;
#pragma unroll
    for (int rr = 0; rr < 8; ++rr)
        Gp[(size_t)(rbase + rr) * B + col] = acc[rr];
}

// ---------------------------------------------------------------------------
// Kernel 3: G = sum of SPLIT partials.
// ---------------------------------------------------------------------------
__global__ void __launch_bounds__(256) reduce_G(const float* __restrict__ Gpart,
                                                float* __restrict__ G) {
    int idx = blockIdx.x * 256 + threadIdx.x;   // 64 blocks * 256 = 16384 = B*B
    float s = 0.f;
#pragma unroll
    for (int k = 0; k < SPLIT; ++k) s += Gpart[(size_t)k * B * B + idx];
    G[idx] = s;
}

// ---------------------------------------------------------------------------
// Kernel 4: scalar loss over positive pairs (i<j, same target).
//   loss_ij = -(G[i,j] - sum(ss_mean)) + 0.5*(lam[i]+lam[j])
// ---------------------------------------------------------------------------
__global__ void __launch_bounds__(256) finalize(const float* __restrict__ G,
                                                const float* __restrict__ lam,
                                                const int* __restrict__ targets,
                                                const float* __restrict__ ss_mean,
                                                float* __restrict__ out) {
    __shared__ float redS[256];
    __shared__ float redC[256];
    __shared__ float smsum;
    int tid = threadIdx.x;
    if (tid == 0) {
        float s = 0.f;
        for (int i = 0; i < P * P; ++i) s += ss_mean[i];
        smsum = s;
    }
    __syncthreads();
    float msum = smsum;
    float ls = 0.f, lc = 0.f;
    for (int idx = tid; idx < B * B; idx += 256) {
        int i = idx / B, j = idx % B;
        if (i < j && targets[i] == targets[j]) {
            ls += -(G[idx] - msum) + 0.5f * (lam[i] + lam[j]);
            lc += 1.0f;
        }
    }
    redS[tid] = ls;
    redC[tid] = lc;
    __syncthreads();
    for (int s = 128; s > 0; s >>= 1) {
        if (tid < s) { redS[tid] += redS[tid + s]; redC[tid] += redC[tid + s]; }
        __syncthreads();
    }
    if (tid == 0) out[0] = redS[0] / redC[0];
}

// ---------------------------------------------------------------------------
extern "C" void kernel_launch(void* const* d_in, const int* in_sizes, int n_in,
                              void* d_out, int out_size, void* d_ws, size_t ws_size,
                              hipStream_t stream) {
    (void)in_sizes; (void)n_in; (void)out_size; (void)ws_size;
    const float* x       = (const float*)d_in[0];   // (B, D, P, 1) f32
    const float* ss_mean = (const float*)d_in[1];   // (P, P) f32
    const int*   targets = (const int*)d_in[2];     // (B,) i32
    // d_in[3] = matching_inputs, unused on the default path

    float* F     = (float*)d_ws;                       // B*K floats (~21 MB)
    float* Gpart = F + (size_t)B * K;                  // SPLIT*B*B floats (1 MB)
    float* G     = Gpart + (size_t)SPLIT * B * B;      // B*B floats
    float* lam   = G + (size_t)B * B;                  // B floats

    build_features<<<B, 256, 0, stream>>>(x, F, lam);
    gram_wmma<<<(TILES * SPLIT) / 4, 128, 0, stream>>>(F, Gpart);
    reduce_G<<<(B * B) / 256, 256, 0, stream>>>(Gpart, G);
    finalize<<<1, 256, 0, stream>>>(G, lam, targets, ss_mean, (float*)d_out);
}